// _CIN_41575283425691
// MI455X (gfx1250) — compile-verified
//
#include <hip/hip_runtime.h>

typedef float v2f __attribute__((ext_vector_type(2)));
typedef float v8f __attribute__((ext_vector_type(8)));

// One block: TWO batches (blockIdx.x -> b0=2*blockIdx.x), 64-row output chunk
// (blockIdx.y in 0..3). 128 threads = 4 waves (wave32); wave w owns rows
// [chunk*64 + w*16, +16) x 64 cols for BOTH batches (A-fragment reuse: one
// W load feeds 8 WMMAs -> halves L2 traffic for the shared weight matrix).
// GEMM: out(200x64) = W(200x(40*NH)) @ P((40*NH)x64), P[(m,n),d] = X[m,d]*H[n,d].
template <int NH>
__global__ __launch_bounds__(128) void fused_layer_kernel(
    const float* __restrict__ x,     // (512, 40, 64)
    const float* __restrict__ Hsrc,  // (512, NH, 64)
    const float* __restrict__ W,     // (200, 40*NH) row-major
    const float* __restrict__ bias,  // (200)
    float* __restrict__ hid_out,     // (512, 100, 64) or nullptr
    float* __restrict__ out,         // (512, 400)
    int out_base, int direct_start) {
  constexpr int KTOT = 40 * NH;
  const int b0    = blockIdx.x * 2;
  const int chunk = blockIdx.y;          // rows chunk*64 .. chunk*64+63
  const int tid   = threadIdx.x;         // 0..127
  const int wave  = tid >> 5;            // 0..3
  const int lane  = tid & 31;
  const int l16   = lane & 15;
  const int half  = lane >> 4;           // 0 or 1
  const int koff  = half << 1;           // K offset inside a 4-wide k-step

  // dynamic LDS layout (floats): Xs0|Xs1|Hs0|Hs1|Bs|Os
  extern __shared__ __align__(16) float smem[];
  float* Xs0 = smem;                 // 40*64
  float* Xs1 = smem + 2560;          // 40*64
  float* Hs0 = smem + 5120;          // NH*64
  float* Hs1 = Hs0 + NH * 64;        // NH*64
  float* Bs  = Hs1 + NH * 64;        // 64
  float* Os  = Bs + 64;              // 64*64

  // ---- stage X, H (both batches), bias chunk into LDS ----
  {
    const float4* xa = (const float4*)(x + (size_t)b0 * (40 * 64));
    const float4* xb = (const float4*)(x + (size_t)(b0 + 1) * (40 * 64));
    for (int i = tid; i < 40 * 16; i += 128) {
      ((float4*)Xs0)[i] = xa[i];
      ((float4*)Xs1)[i] = xb[i];
    }
    const float4* ha = (const float4*)(Hsrc + (size_t)b0 * (NH * 64));
    const float4* hb = (const float4*)(Hsrc + (size_t)(b0 + 1) * (NH * 64));
    for (int i = tid; i < NH * 16; i += 128) {
      ((float4*)Hs0)[i] = ha[i];
      ((float4*)Hs1)[i] = hb[i];
    }
    if (tid < 64) {
      int r = chunk * 64 + tid;
      Bs[tid] = bias[r < 200 ? r : 199];
    }
  }
  __syncthreads();

  // ---- A-fragment row pointer (clamped for the padded tail rows) ----
  const int m_base = chunk * 64 + wave * 16;
  int arow = m_base + l16;
  if (arow > 199) arow = 199;            // safe garbage, masked at store
  const float* wrow = W + (size_t)arow * KTOT;

  v8f acc[2][4];
#pragma unroll
  for (int bi = 0; bi < 2; ++bi)
#pragma unroll
    for (int t = 0; t < 4; ++t) acc[bi][t] = (v8f){};

  // Software-pipelined A fragment: a_cur holds k-step k0, load k0+4 early.
  v2f a_cur = *(const v2f*)(wrow + koff);

  // ---- main k-loop: k = m*NH + n, NH % 4 == 0 so steps never cross m ----
  for (int m = 0; m < 40; ++m) {
    const float* xm0 = Xs0 + m * 64;
    const float* xm1 = Xs1 + m * 64;
    const float xa0 = xm0[l16 + 0],  xb0 = xm1[l16 + 0];
    const float xa1 = xm0[l16 + 16], xb1 = xm1[l16 + 16];
    const float xa2 = xm0[l16 + 32], xb2 = xm1[l16 + 32];
    const float xa3 = xm0[l16 + 48], xb3 = xm1[l16 + 48];
    const int kbase = m * NH;
    // pull next m-row chunk of W toward the WGP while the matrix pipe is busy
    __builtin_prefetch(wrow + kbase + NH + koff, 0, 1);
#pragma unroll 5
    for (int n0 = 0; n0 < NH; n0 += 4) {
      const int k0 = kbase + n0;
      // prefetch next k-step's A frag into registers (clamped at the end)
      int kn = k0 + 4;
      if (kn > KTOT - 4) kn = KTOT - 4;
      v2f a_nxt = *(const v2f*)(wrow + kn + koff);
      const int nr = n0 + koff;
      // batch 0 B frags of P (d = l16 + 16*t)
      const float* g0 = Hs0 + nr * 64;
      const float* g1 = g0 + 64;
      v2f p00 = {xa0 * g0[l16 + 0],  xa0 * g1[l16 + 0]};
      v2f p01 = {xa1 * g0[l16 + 16], xa1 * g1[l16 + 16]};
      v2f p02 = {xa2 * g0[l16 + 32], xa2 * g1[l16 + 32]};
      v2f p03 = {xa3 * g0[l16 + 48], xa3 * g1[l16 + 48]};
      acc[0][0] = __builtin_amdgcn_wmma_f32_16x16x4_f32(false, a_cur, false, p00, (short)0, acc[0][0], false, false);
      acc[0][1] = __builtin_amdgcn_wmma_f32_16x16x4_f32(false, a_cur, false, p01, (short)0, acc[0][1], false, false);
      acc[0][2] = __builtin_amdgcn_wmma_f32_16x16x4_f32(false, a_cur, false, p02, (short)0, acc[0][2], false, false);
      acc[0][3] = __builtin_amdgcn_wmma_f32_16x16x4_f32(false, a_cur, false, p03, (short)0, acc[0][3], false, false);
      // batch 1 B frags (reuse the same A registers)
      const float* q0 = Hs1 + nr * 64;
      const float* q1 = q0 + 64;
      v2f p10 = {xb0 * q0[l16 + 0],  xb0 * q1[l16 + 0]};
      v2f p11 = {xb1 * q0[l16 + 16], xb1 * q1[l16 + 16]};
      v2f p12 = {xb2 * q0[l16 + 32], xb2 * q1[l16 + 32]};
      v2f p13 = {xb3 * q0[l16 + 48], xb3 * q1[l16 + 48]};
      acc[1][0] = __builtin_amdgcn_wmma_f32_16x16x4_f32(false, a_cur, false, p10, (short)0, acc[1][0], false, false);
      acc[1][1] = __builtin_amdgcn_wmma_f32_16x16x4_f32(false, a_cur, false, p11, (short)0, acc[1][1], false, false);
      acc[1][2] = __builtin_amdgcn_wmma_f32_16x16x4_f32(false, a_cur, false, p12, (short)0, acc[1][2], false, false);
      acc[1][3] = __builtin_amdgcn_wmma_f32_16x16x4_f32(false, a_cur, false, p13, (short)0, acc[1][3], false, false);
      a_cur = a_nxt;
    }
  }

  // ---- per-batch epilogue: bias+relu, stage to LDS, hidden write, row sums ----
#pragma unroll
  for (int bi = 0; bi < 2; ++bi) {
    const int bcur = b0 + bi;
    __syncthreads();  // Os free (k-loop done / previous batch consumed)
#pragma unroll
    for (int j = 0; j < 8; ++j) {
      const int rl = wave * 16 + j + 8 * half;  // local row in chunk
      const float bb = Bs[rl];
#pragma unroll
      for (int t = 0; t < 4; ++t) {
        float v = acc[bi][t][j] + bb;
        Os[rl * 64 + l16 + 16 * t] = v > 0.f ? v : 0.f;
      }
    }
    __syncthreads();

    // hidden write: global rows o in [0,100) feed the next layer
    if (hid_out != nullptr) {
      float* hob = hid_out + (size_t)bcur * (100 * 64);
      for (int i = tid; i < 64 * 16; i += 128) {  // float4 granules over Os
        const int rl = i >> 4;
        const int o = chunk * 64 + rl;
        if (o < 100)
          ((float4*)(hob + (size_t)o * 64))[i & 15] = ((const float4*)Os)[i];
      }
    }

    // row sums over d for the "direct" rows -> d_out
    if (tid < 64) {
      const int o = chunk * 64 + tid;
      if (o < 200 && o >= direct_start) {
        float s = 0.f;
        const float* row = Os + tid * 64;
#pragma unroll 8
        for (int d2 = 0; d2 < 64; ++d2) s += row[d2];
        out[(size_t)bcur * 400 + out_base + (o - direct_start)] = s;
      }
    }
  }
}

extern "C" void kernel_launch(void* const* d_in, const int* in_sizes, int n_in,
                              void* d_out, int out_size, void* d_ws, size_t ws_size,
                              hipStream_t stream) {
  (void)in_sizes; (void)n_in; (void)out_size; (void)ws_size;
  const float* x  = (const float*)d_in[0];
  const float* w1 = (const float*)d_in[1];
  const float* b1 = (const float*)d_in[2];
  const float* w2 = (const float*)d_in[3];
  const float* b2 = (const float*)d_in[4];
  const float* w3 = (const float*)d_in[5];
  const float* b3 = (const float*)d_in[6];
  float* out  = (float*)d_out;
  float* hid1 = (float*)d_ws;                       // 512*100*64 f32 = 13.1 MB
  float* hid2 = hid1 + (size_t)512 * 100 * 64;      // second 13.1 MB

  dim3 grid(256, 4), block(128);
  const size_t smem40  = (size_t)(5120 + 2 * 40 * 64 + 64 + 4096) * 4;   // 57,600 B
  const size_t smem100 = (size_t)(5120 + 2 * 100 * 64 + 64 + 4096) * 4;  // 88,320 B
  // layer 1: hidden = x (NH=40); direct rows 100..199 -> out cols 0..99
  fused_layer_kernel<40><<<grid, block, smem40, stream>>>(x, x, w1, b1, hid1, out, 0, 100);
  // layer 2: NH=100; direct rows 100..199 -> out cols 100..199
  fused_layer_kernel<100><<<grid, block, smem100, stream>>>(x, hid1, w2, b2, hid2, out, 100, 100);
  // layer 3: NH=100; all rows 0..199 -> out cols 200..399
  fused_layer_kernel<100><<<grid, block, smem100, stream>>>(x, hid2, w3, b3, nullptr, out, 200, 0);
}